// GraphAttentionLayer_68247030333748
// MI455X (gfx1250) — compile-verified
//
#include <hip/hip_runtime.h>
#include <math.h>

#define N_NODES 50000
#define N_EDGES 800000
#define IN_F 256
#define OUT_F 64
#define HEADS 4
#define NEG_SLOPE 0.2f

typedef __attribute__((ext_vector_type(2))) float v2f;
typedef __attribute__((ext_vector_type(8))) float v8f;

// ---- workspace layout (in floats) ----
#define SZ_H   (HEADS * N_NODES * OUT_F)       // 12,800,000 : h = x@W, [H][N][64]
#define OFF_H    0
#define OFF_ASRC (OFF_H + SZ_H)                // attn_src [H][N]
#define OFF_ADST (OFF_ASRC + HEADS * N_NODES)  // attn_dst [H][N]
#define OFF_DEN  (OFF_ADST + HEADS * N_NODES)  // denom    [H][N]
#define OFF_E    (OFF_DEN + HEADS * N_NODES)   // e / e_exp [H][E]
#define OFF_MAX  (OFF_E + HEADS * N_EDGES)     // global max scalar

// ------------------------------------------------------------------
// init: zero output + denominators, set global max = -inf
// ------------------------------------------------------------------
__global__ void k_init(float* __restrict__ out, float* __restrict__ denom,
                       float* __restrict__ maxv) {
    int i = blockIdx.x * blockDim.x + threadIdx.x;
    if (i < N_NODES * HEADS * OUT_F) out[i] = 0.0f;
    if (i < HEADS * N_NODES) denom[i] = 0.0f;
    if (i == 0) *maxv = -__builtin_inff();
}

// ------------------------------------------------------------------
// GEMM: h[head][node][o] = sum_k x[node][k] * W[head][k][o]
// M=50000 (3125 tiles of 16), K=256, N=256 (16 n-tiles; head = n>>6)
// One block = 8 waves; x tile (16x256 f32) staged in LDS; each wave
// owns n-tiles {w, w+8} -> two v8f accumulators, 2 WMMA per 4-K step.
// ------------------------------------------------------------------
__global__ __launch_bounds__(256) void k_gemm(const float* __restrict__ x,
                                              const float* __restrict__ W,
                                              float* __restrict__ hbuf) {
    __shared__ float xs[16 * IN_F];
    const int m0 = blockIdx.x * 16;
    {   // cooperative coalesced tile load: 1024 float4s
        const float4* xg = (const float4*)(x + (size_t)m0 * IN_F);
        float4* xl = (float4*)xs;
        for (int t = threadIdx.x; t < (16 * IN_F) / 4; t += 256) xl[t] = xg[t];
    }
    __syncthreads();

    const int wave  = threadIdx.x >> 5;
    const int lane  = threadIdx.x & 31;
    const int lhalf = lane >> 4;     // 0: lanes 0-15 (K 0/1), 1: lanes 16-31 (K 2/3)
    const int l15   = lane & 15;     // M index for A, N index for B/C/D

    const int n0 = wave * 16;        // first n-tile column base
    const int n1 = (wave + 8) * 16;  // second n-tile column base
    const int head0 = n0 >> 6, o0 = n0 & 63;
    const int head1 = n1 >> 6, o1 = n1 & 63;

    v8f c0 = {}; v8f c1 = {};
    for (int kk = 0; kk < IN_F; kk += 4) {
        const int ka = kk + 2 * lhalf;
        // A fragment (16x4): lane l15 = row m, two consecutive K values
        v2f a;
        a.x = xs[l15 * IN_F + ka];
        a.y = xs[l15 * IN_F + ka + 1];
        // B fragment (4x16): B[k][n] = W[head*IN_F*OUT_F + k*OUT_F + o]
        v2f b0, b1;
        {
            const float* Wp = W + (size_t)head0 * (IN_F * OUT_F) + (size_t)ka * OUT_F + (o0 + l15);
            b0.x = Wp[0];
            b0.y = Wp[OUT_F];
        }
        {
            const float* Wp = W + (size_t)head1 * (IN_F * OUT_F) + (size_t)ka * OUT_F + (o1 + l15);
            b1.x = Wp[0];
            b1.y = Wp[OUT_F];
        }
        c0 = __builtin_amdgcn_wmma_f32_16x16x4_f32(false, a, false, b0, (short)0, c0, false, false);
        c1 = __builtin_amdgcn_wmma_f32_16x16x4_f32(false, a, false, b1, (short)0, c1, false, false);
    }

    // D layout: VGPR r, lanes 0-15 -> M=r, lanes 16-31 -> M=r+8; N = l15
    #pragma unroll
    for (int r = 0; r < 8; ++r) {
        const int node = m0 + (lhalf ? (r + 8) : r);
        hbuf[((size_t)head0 * N_NODES + node) * OUT_F + (o0 + l15)] = c0[r];
        hbuf[((size_t)head1 * N_NODES + node) * OUT_F + (o1 + l15)] = c1[r];
    }
}

// ------------------------------------------------------------------
// attention logits: wave per (node, head); 64-wide dot via shuffle
// ------------------------------------------------------------------
__global__ __launch_bounds__(256) void k_attn(const float* __restrict__ hbuf,
                                              const float* __restrict__ a_src,
                                              const float* __restrict__ a_dst,
                                              float* __restrict__ attn_s,
                                              float* __restrict__ attn_d) {
    const int wid = (blockIdx.x * blockDim.x + threadIdx.x) >> 5;
    if (wid >= N_NODES * HEADS) return;
    const int lane = threadIdx.x & 31;
    const int node = wid >> 2;
    const int head = wid & 3;
    const float2 hv = ((const float2*)(hbuf + ((size_t)head * N_NODES + node) * OUT_F))[lane];
    const float2 as = ((const float2*)(a_src + head * OUT_F))[lane];
    const float2 ad = ((const float2*)(a_dst + head * OUT_F))[lane];
    float ps = hv.x * as.x + hv.y * as.y;
    float pd = hv.x * ad.x + hv.y * ad.y;
    #pragma unroll
    for (int off = 16; off > 0; off >>= 1) {
        ps += __shfl_down(ps, off, 32);
        pd += __shfl_down(pd, off, 32);
    }
    if (lane == 0) {
        attn_s[head * N_NODES + node] = ps;
        attn_d[head * N_NODES + node] = pd;
    }
}

// IEEE-ordered float atomic max via int/uint punning
__device__ __forceinline__ void atomicMaxFloat(float* addr, float val) {
    if (!(val == val)) return;
    if (val >= 0.0f) atomicMax((int*)addr, __float_as_int(val));
    else             atomicMin((unsigned int*)addr, __float_as_uint(val));
}

// ------------------------------------------------------------------
// per-edge raw scores + global max (block reduce -> one atomic)
// ------------------------------------------------------------------
__global__ __launch_bounds__(256) void k_escore(const int* __restrict__ src_idx,
                                                const int* __restrict__ dst_idx,
                                                const float* __restrict__ ew,
                                                const float* __restrict__ attn_s,
                                                const float* __restrict__ attn_d,
                                                float* __restrict__ ebuf,
                                                float* __restrict__ maxv) {
    const int e = blockIdx.x * blockDim.x + threadIdx.x;
    float lmax = -__builtin_inff();
    if (e < N_EDGES) {
        const int s = src_idx[e], d = dst_idx[e];
        const float w = ew[e];
        #pragma unroll
        for (int h = 0; h < HEADS; ++h) {
            float v = attn_s[h * N_NODES + s] + attn_d[h * N_NODES + d];
            v = (v >= 0.0f) ? v : NEG_SLOPE * v;   // leaky relu
            v *= w;
            ebuf[(size_t)h * N_EDGES + e] = v;
            lmax = fmaxf(lmax, v);
        }
    }
    __shared__ float red[256];
    red[threadIdx.x] = lmax;
    __syncthreads();
    for (int s = 128; s > 0; s >>= 1) {
        if (threadIdx.x < s) red[threadIdx.x] = fmaxf(red[threadIdx.x], red[threadIdx.x + s]);
        __syncthreads();
    }
    if (threadIdx.x == 0) atomicMaxFloat(maxv, red[0]);
}

// ------------------------------------------------------------------
// e_exp = exp(e - max); denominator segment-sum per (head, dst)
// ------------------------------------------------------------------
__global__ __launch_bounds__(256) void k_expdenom(const int* __restrict__ dst_idx,
                                                  float* __restrict__ ebuf,
                                                  const float* __restrict__ maxv,
                                                  float* __restrict__ denom) {
    const int i = blockIdx.x * blockDim.x + threadIdx.x;
    if (i >= HEADS * N_EDGES) return;
    const int h = i / N_EDGES;
    const int e = i - h * N_EDGES;
    const float val = expf(ebuf[i] - *maxv);
    ebuf[i] = val;
    atomicAdd(&denom[h * N_NODES + dst_idx[e]], val);
}

// ------------------------------------------------------------------
// aggregation: wave per edge; lane l covers head=l>>3, o=(l&7)*8..+7
// coalesced float4 gathers of h[src], contiguous f32 atomics into out
// ------------------------------------------------------------------
__global__ __launch_bounds__(256) void k_aggregate(const int* __restrict__ src_idx,
                                                   const int* __restrict__ dst_idx,
                                                   const float* __restrict__ ebuf,
                                                   const float* __restrict__ denom,
                                                   const float* __restrict__ hbuf,
                                                   float* __restrict__ out) {
    const int wid = (blockIdx.x * blockDim.x + threadIdx.x) >> 5;
    if (wid >= N_EDGES) return;
    const int lane = threadIdx.x & 31;
    const int head = lane >> 3;
    const int ob   = (lane & 7) * 8;
    const int s = src_idx[wid];
    const int d = dst_idx[wid];
    const float alpha = ebuf[(size_t)head * N_EDGES + wid] /
                        (denom[head * N_NODES + d] + 1e-10f);
    const float4* hp = (const float4*)(hbuf + ((size_t)head * N_NODES + s) * OUT_F + ob);
    const float4 m0 = hp[0], m1 = hp[1];
    float* op = out + (size_t)d * (HEADS * OUT_F) + head * OUT_F + ob;
    atomicAdd(op + 0, m0.x * alpha);
    atomicAdd(op + 1, m0.y * alpha);
    atomicAdd(op + 2, m0.z * alpha);
    atomicAdd(op + 3, m0.w * alpha);
    atomicAdd(op + 4, m1.x * alpha);
    atomicAdd(op + 5, m1.y * alpha);
    atomicAdd(op + 6, m1.z * alpha);
    atomicAdd(op + 7, m1.w * alpha);
}

extern "C" void kernel_launch(void* const* d_in, const int* in_sizes, int n_in,
                              void* d_out, int out_size, void* d_ws, size_t ws_size,
                              hipStream_t stream) {
    const float* x     = (const float*)d_in[0];
    const int*   eidx  = (const int*)d_in[1];
    const float* ew    = (const float*)d_in[2];
    const float* W     = (const float*)d_in[3];
    const float* a_src = (const float*)d_in[4];
    const float* a_dst = (const float*)d_in[5];
    const int* src_idx = eidx;             // edge_index row 0
    const int* dst_idx = eidx + N_EDGES;   // edge_index row 1

    float* ws     = (float*)d_ws;
    float* hbuf   = ws + OFF_H;
    float* attn_s = ws + OFF_ASRC;
    float* attn_d = ws + OFF_ADST;
    float* denom  = ws + OFF_DEN;
    float* ebuf   = ws + OFF_E;
    float* maxv   = ws + OFF_MAX;
    float* out    = (float*)d_out;

    k_init<<<(N_NODES * HEADS * OUT_F + 255) / 256, 256, 0, stream>>>(out, denom, maxv);
    k_gemm<<<N_NODES / 16, 256, 0, stream>>>(x, W, hbuf);
    k_attn<<<(N_NODES * HEADS * 32 + 255) / 256, 256, 0, stream>>>(hbuf, a_src, a_dst, attn_s, attn_d);
    k_escore<<<(N_EDGES + 255) / 256, 256, 0, stream>>>(src_idx, dst_idx, ew, attn_s, attn_d, ebuf, maxv);
    k_expdenom<<<(HEADS * N_EDGES + 255) / 256, 256, 0, stream>>>(dst_idx, ebuf, maxv, denom);
    k_aggregate<<<(N_EDGES * 32 + 255) / 256, 256, 0, stream>>>(src_idx, dst_idx, ebuf, denom, hbuf, out);
}